// ProjectionLayer_72756745994440
// MI455X (gfx1250) — compile-verified
//
#include <hip/hip_runtime.h>
#include <stdint.h>

typedef float    v4f __attribute__((ext_vector_type(4)));
typedef uint32_t v4u __attribute__((ext_vector_type(4)));
typedef int      v4i __attribute__((ext_vector_type(4)));
typedef int      v8i __attribute__((ext_vector_type(8)));

// Transposed workspace layout (element offsets):
//   scale0: [56*56, 64]  at      0   (200704 elems)
//   scale1: [28*28,128]  at 200704   (100352 elems)
//   scale2: [14*14,256]  at 301056   ( 50176 elems)
//   scale3: [ 7* 7,512]  at 351232   ( 25088 elems)
#define T1_OFF 200704
#define T2_OFF 301056
#define T3_OFF 351232
#define WS_ELEMS 376320

#define LDS2_F 50176              // feat2T floats staged at lds[0]
#define LDS3_F 25088              // feat3T floats staged at lds[LDS2_F]
#define LDS_BYTES ((LDS2_F + LDS3_F) * 4)   // 301056 B <= 320KB/WGP

// 5-arg TDM builtin exists on ROCm 7.2 (clang-22); the therock toolchain ships
// the TDM header and a different (6-arg) builtin -> use async loads there.
#if defined(__has_builtin)
#  if __has_builtin(__builtin_amdgcn_tensor_load_to_lds) && \
      !__has_include(<hip/amd_detail/amd_gfx1250_TDM.h>)
#    define USE_TDM_5ARG 1
#  endif
#endif

// ---------------------------------------------------------------------------
// Kernel 1: transpose feat[batch] from [C,H,W] to [H*W, C] for all 4 scales.
// ---------------------------------------------------------------------------
__global__ __launch_bounds__(256) void k_proj_transpose(
    const float* __restrict__ f0, const float* __restrict__ f1,
    const float* __restrict__ f2, const float* __restrict__ f3,
    const int* __restrict__ pbatch, float* __restrict__ ws) {
  int t = blockIdx.x * 256 + threadIdx.x;
  if (t >= WS_ELEMS) return;
  int b = *pbatch;
  const float* f; int local, clog, hw_sz;
  if (t < T1_OFF)      { f = f0 + (size_t)b * 200704; local = t;          clog = 6; hw_sz = 3136; }
  else if (t < T2_OFF) { f = f1 + (size_t)b * 100352; local = t - T1_OFF; clog = 7; hw_sz = 784;  }
  else if (t < T3_OFF) { f = f2 + (size_t)b * 50176;  local = t - T2_OFF; clog = 8; hw_sz = 196;  }
  else                 { f = f3 + (size_t)b * 25088;  local = t - T3_OFF; clog = 9; hw_sz = 49;   }
  int c  = local & ((1 << clog) - 1);
  int hw = local >> clog;
  ws[t] = f[(size_t)c * hw_sz + hw];
}

// ---------------------------------------------------------------------------
// Projection math — bit-exact vs reference (224/S is an exact power of two,
// and the reference's int-truncation makes w21=w12=w22 == 0, w11 in {0,1}).
// ---------------------------------------------------------------------------
__device__ __forceinline__ void proj_scale(float h, float w, float inv, int S,
                                           int& cell, float& wt) {
  float x = h * inv, y = w * inv;
  int x1 = (int)floorf(x);
  int y1 = (int)floorf(y);
  int x2 = (int)ceilf(x);  if (x2 > S - 1) x2 = S - 1;
  int y2 = (int)ceilf(y);  if (y2 > S - 1) y2 = S - 1;
  cell = x1 * S + y1;
  wt   = (float)((x2 - x1) * (y2 - y1));
}

__device__ __forceinline__ void point_hw(const float* __restrict__ pts, int n,
                                         float& h, float& w) {
  float px = pts[3 * n + 0], py = pts[3 * n + 1], pz = pts[3 * n + 2];
  h = 248.0f * (py / pz) + 111.5f;
  w = 248.0f * (px / (-pz)) + 111.5f;
  h = fminf(fmaxf(h, 0.0f), 223.0f);
  w = fminf(fmaxf(w, 0.0f), 223.0f);
}

// ---------------------------------------------------------------------------
// Kernel 2: main gather. 512 threads = 16 waves; each wave owns one point at a
// time. feat2T is staged to LDS with async global->LDS b128 (ASYNCcnt); feat3T
// with one Tensor Data Mover descriptor (TENSORcnt). 80% of gather reads then
// come from LDS; scales 0/1 read the L2-resident workspace with b128 loads.
// Output rows stream out with non-temporal b128 stores.
// ---------------------------------------------------------------------------
__global__ __launch_bounds__(512) void k_proj_gather(
    const float* __restrict__ pts, const float* __restrict__ ws,
    float* __restrict__ out, int N) {
  extern __shared__ float lds[];  // [0,50176): feat2T  [50176,75264): feat3T

  uint32_t lbase = (uint32_t)(uintptr_t)(void*)lds;  // low 32b = LDS offset

  // --- feat3T (49 rows x 2048 B) via TDM: one instruction per block ---------
#ifdef USE_TDM_5ARG
  if (__builtin_amdgcn_readfirstlane((int)(threadIdx.x >> 5)) == 0) {
    uint64_t ga = (uint64_t)(uintptr_t)(const void*)(ws + T3_OFF);
    v4u g0;
    g0.x = 1u;                                   // count=1, user mode
    g0.y = lbase + (uint32_t)(LDS2_F * 4);       // lds_addr (bytes)
    g0.z = (uint32_t)(ga & 0xffffffffu);         // global_addr[31:0]
    g0.w = (uint32_t)((ga >> 32) & 0x01ffffffu)  // global_addr[56:32]
           | (2u << 30);                         // type=2 (image)
    v8i g1;
    g1[0] = 0x20000;                             // data_size=4B, mask=0
    g1[1] = (int)(512u << 16);                   // tensor_dim0[15:0]=512
    g1[2] = (int)(49u << 16);                    // dim0[31:16]=0, dim1[15:0]=49
    g1[3] = (int)(512u << 16);                   // dim1[31:16]=0, tile_dim0=512
    g1[4] = 49;                                  // tile_dim1=49, tile_dim2=0
    g1[5] = 512;                                 // tensor_dim0_stride[31:0]
    g1[6] = 0;                                   // stride0[47:32], stride1[15:0]
    g1[7] = 0;
    v4i gz = {0, 0, 0, 0};
    __builtin_amdgcn_tensor_load_to_lds(g0, g1, gz, gz, 0);
    __builtin_amdgcn_s_wait_tensorcnt(0);
  }
#else
  {
    uint64_t sbase = (uint64_t)(uintptr_t)(const void*)(ws + T3_OFF);
    uint32_t lb3 = lbase + (uint32_t)(LDS2_F * 4);
    for (int i = threadIdx.x; i < LDS3_F / 4; i += 512) {
      uint32_t voff = (uint32_t)(i * 16);
      asm volatile("global_load_async_to_lds_b128 %0, %1, %2"
                   :: "v"(lb3 + voff), "v"(voff), "s"(sbase) : "memory");
    }
  }
#endif

  // --- feat2T (200704 B) via async global->LDS b128 -------------------------
  {
    uint64_t sbase = (uint64_t)(uintptr_t)(const void*)(ws + T2_OFF);
    for (int i = threadIdx.x; i < LDS2_F / 4; i += 512) {
      uint32_t voff = (uint32_t)(i * 16);
      asm volatile("global_load_async_to_lds_b128 %0, %1, %2"
                   :: "v"(lbase + voff), "v"(voff), "s"(sbase) : "memory");
    }
    asm volatile("s_wait_asynccnt 0" ::: "memory");
  }
  __syncthreads();

  const int lane = threadIdx.x & 31;
  const int wid  = threadIdx.x >> 5;           // 0..15
  const int base = blockIdx.x * 512;

  for (int k = 0; k < 32; ++k) {
    int n = base + k * 16 + wid;
    if (n >= N) continue;

    float h, w;
    point_hw(pts, n, h, w);

    int cl0, cl1, cl2, cl3; float w0, w1, w2, w3;
    proj_scale(h, w, 0.25f,    56, cl0, w0);
    proj_scale(h, w, 0.125f,   28, cl1, w1);
    proj_scale(h, w, 0.0625f,  14, cl2, w2);
    proj_scale(h, w, 0.03125f,  7, cl3, w3);
    int c0 = cl0 * 64, c1 = cl1 * 128, c2 = cl2 * 256, c3 = cl3 * 512;

    size_t ob = (size_t)n * 960;
    // 960 channels = 240 float4 slots: s0 [0,16) s1 [16,48) s2 [48,112) s3 [112,240)
#pragma unroll
    for (int j = 0; j < 8; ++j) {
      int idx = j * 32 + lane;
      if (idx < 240) {
        int gc = idx << 2;
        v4f v; float wt;
        if (idx < 16)       { v = *(const v4f*)(ws + (c0 + gc));                 wt = w0; }
        else if (idx < 48)  { v = *(const v4f*)(ws + (T1_OFF + c1 + (gc - 64))); wt = w1; }
        else if (idx < 112) { v = *(const v4f*)(lds + (c2 + (gc - 192)));        wt = w2; }
        else                { v = *(const v4f*)(lds + (LDS2_F + c3 + (gc - 448))); wt = w3; }
        v4f o = v * wt;
        __builtin_nontemporal_store(o, (v4f*)(out + ob + gc));
      }
    }
  }
}

// ---------------------------------------------------------------------------
// Fallback (only if ws too small): gather straight from [C,H,W] layout.
// ---------------------------------------------------------------------------
__global__ __launch_bounds__(256) void k_proj_gather_direct(
    const float* __restrict__ pts,
    const float* __restrict__ f0, const float* __restrict__ f1,
    const float* __restrict__ f2, const float* __restrict__ f3,
    const int* __restrict__ pbatch, float* __restrict__ out, int N) {
  int b = *pbatch;
  const float* F0 = f0 + (size_t)b * 200704;
  const float* F1 = f1 + (size_t)b * 100352;
  const float* F2 = f2 + (size_t)b * 50176;
  const float* F3 = f3 + (size_t)b * 25088;

  int lane = threadIdx.x & 31, wid = threadIdx.x >> 5;
  int n = blockIdx.x * 8 + wid;
  if (n >= N) return;

  float h, w;
  point_hw(pts, n, h, w);
  int cl0, cl1, cl2, cl3; float w0, w1, w2, w3;
  proj_scale(h, w, 0.25f,    56, cl0, w0);
  proj_scale(h, w, 0.125f,   28, cl1, w1);
  proj_scale(h, w, 0.0625f,  14, cl2, w2);
  proj_scale(h, w, 0.03125f,  7, cl3, w3);

  size_t ob = (size_t)n * 960;
  for (int idx = lane; idx < 960; idx += 32) {
    const float* F; int c, hw_sz, cell; float wt;
    if (idx < 64)       { F = F0; c = idx;       hw_sz = 3136; cell = cl0; wt = w0; }
    else if (idx < 192) { F = F1; c = idx - 64;  hw_sz = 784;  cell = cl1; wt = w1; }
    else if (idx < 448) { F = F2; c = idx - 192; hw_sz = 196;  cell = cl2; wt = w2; }
    else                { F = F3; c = idx - 448; hw_sz = 49;   cell = cl3; wt = w3; }
    out[ob + idx] = wt * F[(size_t)c * hw_sz + cell];
  }
}

// ---------------------------------------------------------------------------
extern "C" void kernel_launch(void* const* d_in, const int* in_sizes, int n_in,
                              void* d_out, int out_size, void* d_ws, size_t ws_size,
                              hipStream_t stream) {
  const float* f0  = (const float*)d_in[0];
  const float* f1  = (const float*)d_in[1];
  const float* f2  = (const float*)d_in[2];
  const float* f3  = (const float*)d_in[3];
  const float* pts = (const float*)d_in[4];
  const int* pbatch = (const int*)d_in[5];
  float* out = (float*)d_out;
  int N = in_sizes[4] / 3;  // 131072

  if (ws_size >= (size_t)WS_ELEMS * sizeof(float)) {
    float* ws = (float*)d_ws;
    k_proj_transpose<<<(WS_ELEMS + 255) / 256, 256, 0, stream>>>(f0, f1, f2, f3,
                                                                pbatch, ws);
    (void)hipFuncSetAttribute((const void*)k_proj_gather,
                              hipFuncAttributeMaxDynamicSharedMemorySize,
                              LDS_BYTES);
    int nblocks = (N + 511) / 512;
    k_proj_gather<<<nblocks, 512, LDS_BYTES, stream>>>(pts, ws, out, N);
  } else {
    int nblocks = (N + 7) / 8;
    k_proj_gather_direct<<<nblocks, 256, 0, stream>>>(pts, f0, f1, f2, f3,
                                                      pbatch, out, N);
  }
}